// BaselineDNN_4320737100175
// MI455X (gfx1250) — compile-verified
//
#include <hip/hip_runtime.h>

// ---------------------------------------------------------------------------
// BaselineDNN on MI455X (gfx1250):
//   rep  = mean-pool(emb_table[x])            -> bandwidth bound (L2-resident
//                                                120MB table, ~492MB gather)
//   out  = relu(rep@W1 + b1) @ W2 + b2        -> tiny GEMMs, fp32 WMMA
// Phase 1: gather/pool kernel with global_prefetch of next row.
// Phase 2: fused MLP, one wave per 16-row tile, V_WMMA_F32_16X16X4_F32,
//          ReLU staged through LDS between the two GEMMs.
// ---------------------------------------------------------------------------

typedef __attribute__((ext_vector_type(2))) float v2f;
typedef __attribute__((ext_vector_type(8))) float v8f;

#define EMB   300
#define BATCH 2048
#define SEQ   200
#define HID   128
#define OUT_N 20

// ---------------------------------------------------------------------------
// Kernel 1: rep[b, :] = sum_l emb_table[x[b,l], :] / lengths[b]
// One block per batch row. Indices staged in LDS; 256 threads stride the
// 300 columns (1024B + 176B coalesced bursts). Prefetch next gathered row.
// ---------------------------------------------------------------------------
__global__ __launch_bounds__(256) void embed_mean_kernel(
    const int* __restrict__ x, const int* __restrict__ lengths,
    const float* __restrict__ table, float* __restrict__ rep)
{
  __shared__ int s_idx[SEQ];
  const int b = blockIdx.x;
  const int t = threadIdx.x;

  for (int l = t; l < SEQ; l += 256) s_idx[l] = x[b * SEQ + l];
  __syncthreads();

  const int c0 = t;          // t < 256 < 300: always valid
  const int c1 = t + 256;    // valid only for t < 44
  float acc0 = 0.0f, acc1 = 0.0f;

  for (int l = 0; l < SEQ; ++l) {
    const float* row = table + (long)s_idx[l] * EMB;
    // Prefetch next row (speculative, gfx1250 global_prefetch_b8).
    const int lnext = (l + 1 < SEQ) ? (l + 1) : l;
    const float* nrow = table + (long)s_idx[lnext] * EMB;
    __builtin_prefetch(nrow + c0, 0, 3);

    acc0 += row[c0];
    if (c1 < EMB) acc1 += row[c1];
  }

  const float inv = 1.0f / (float)lengths[b];
  rep[(long)b * EMB + c0] = acc0 * inv;
  if (c1 < EMB) rep[(long)b * EMB + c1] = acc1 * inv;
}

// ---------------------------------------------------------------------------
// Kernel 2: out = relu(rep@W1 + b1) @ W2 + b2, fp32 WMMA 16x16x4.
// 16 blocks x 256 threads (8 waves). Wave w handles batch rows
// [blockIdx*128 + w*16, +16).
//
// Fragment layouts per CDNA5 ISA 7.12.2 (32-bit, wave32):
//   A (16x4):  lanes 0-15 -> M=0..15 with K = {0,1} in VGPR{0,1};
//              lanes 16-31 -> same M with K = {2,3}.
//   B (4x16):  lanes 0-15 -> N=0..15 rows K={0,1}; lanes 16-31 rows K={2,3}.
//   C/D 16x16: VGPR j: lanes 0-15 -> M=j, N=lane; lanes 16-31 -> M=j+8.
// ---------------------------------------------------------------------------
__global__ __launch_bounds__(256) void mlp_wmma_kernel(
    const float* __restrict__ rep, const float* __restrict__ W1,
    const float* __restrict__ b1,  const float* __restrict__ W2,
    const float* __restrict__ b2,  float* __restrict__ out)
{
  __shared__ float h_lds[8][16 * HID];   // 8 waves x 16x128 tile = 64 KB

  const int tid  = threadIdx.x;
  const int w    = tid >> 5;        // wave in block: 0..7
  const int lane = tid & 31;
  const int half = lane >> 4;       // 0: K lo pair / M row j, 1: K hi / M row j+8
  const int ln   = lane & 15;       // M (A frag) or N (B,C frags)
  const int kb   = half * 2;        // K base within the 4-wide step
  const int row0 = blockIdx.x * 128 + w * 16;

  // ---- GEMM 1: h(16x128) = rep(16x300) @ W1(300x128) ----------------------
  v8f acc[8] = {};
  const float* arow = rep + (long)(row0 + ln) * EMB;
  for (int k4 = 0; k4 < EMB / 4; ++k4) {
    const int k = k4 * 4 + kb;
    const v2f a = *(const v2f*)(arow + k);       // 8B aligned: k even, EMB even
#pragma unroll
    for (int nt = 0; nt < 8; ++nt) {
      const int col = nt * 16 + ln;
      v2f bf;
      bf.x = W1[(long)k * HID + col];
      bf.y = W1[(long)(k + 1) * HID + col];
      acc[nt] = __builtin_amdgcn_wmma_f32_16x16x4_f32(
          /*neg_a=*/false, a, /*neg_b=*/false, bf,
          /*c_mod=*/(short)0, acc[nt], /*reuse_a=*/false, /*reuse_b=*/false);
    }
  }

  // ---- bias + ReLU, park h tile in LDS ------------------------------------
#pragma unroll
  for (int nt = 0; nt < 8; ++nt) {
    const float bias = b1[nt * 16 + ln];
#pragma unroll
    for (int j = 0; j < 8; ++j) {
      float v = acc[nt][j] + bias;
      v = v > 0.0f ? v : 0.0f;
      h_lds[w][(j + 8 * half) * HID + nt * 16 + ln] = v;
    }
  }
  __syncthreads();   // cross-lane visibility of each wave's h tile

  // ---- GEMM 2: out(16x20) = h(16x128) @ W2(128x20), N padded to 32 --------
  v8f c2[2] = {};
  for (int k4 = 0; k4 < HID / 4; ++k4) {
    const int k = k4 * 4 + kb;
    v2f a;
    a.x = h_lds[w][ln * HID + k];
    a.y = h_lds[w][ln * HID + k + 1];
#pragma unroll
    for (int nt = 0; nt < 2; ++nt) {
      const int col = nt * 16 + ln;
      const bool ok = (col < OUT_N);
      v2f bf;
      bf.x = ok ? W2[k * OUT_N + col]       : 0.0f;
      bf.y = ok ? W2[(k + 1) * OUT_N + col] : 0.0f;
      c2[nt] = __builtin_amdgcn_wmma_f32_16x16x4_f32(
          false, a, false, bf, (short)0, c2[nt], false, false);
    }
  }

  // ---- bias + masked store of 20 valid columns ----------------------------
#pragma unroll
  for (int nt = 0; nt < 2; ++nt) {
    const int col = nt * 16 + ln;
    if (col < OUT_N) {
      const float bias = b2[col];
#pragma unroll
      for (int j = 0; j < 8; ++j) {
        const int row = row0 + j + 8 * half;
        out[(long)row * OUT_N + col] = c2[nt][j] + bias;
      }
    }
  }
}

// ---------------------------------------------------------------------------
// Harness entry point. Inputs (setup_inputs order):
//   0: x [B,L] int32, 1: lengths [B] int32, 2: emb_table [VOCAB,EMB] f32,
//   3: W1 [EMB,H] f32, 4: b1 [H] f32, 5: W2 [H,OUT] f32, 6: b2 [OUT] f32.
// Output: logits [B,OUT] f32. Scratch: rep [B,EMB] f32 = 2.46 MB in d_ws.
// ---------------------------------------------------------------------------
extern "C" void kernel_launch(void* const* d_in, const int* in_sizes, int n_in,
                              void* d_out, int out_size, void* d_ws, size_t ws_size,
                              hipStream_t stream) {
  const int*   x       = (const int*)  d_in[0];
  const int*   lengths = (const int*)  d_in[1];
  const float* table   = (const float*)d_in[2];
  const float* W1      = (const float*)d_in[3];
  const float* b1      = (const float*)d_in[4];
  const float* W2      = (const float*)d_in[5];
  const float* b2      = (const float*)d_in[6];
  float*       logits  = (float*)d_out;
  float*       rep     = (float*)d_ws;   // [BATCH, EMB] scratch

  embed_mean_kernel<<<BATCH, 256, 0, stream>>>(x, lengths, table, rep);
  mlp_wmma_kernel<<<BATCH / 128, 256, 0, stream>>>(rep, W1, b1, W2, b2, logits);
}